// RaytracedOptics_43044162240549
// MI455X (gfx1250) — compile-verified
//
#include <hip/hip_runtime.h>
#include <math.h>

#define NF 21
#define GHC 8
#define GWC 8
#define PSFH 65
#define PSFW 65
#define RPSF 23
#define HIMG 1024
#define WIMG 1024
#define BB 4
#define CC 3
#define NPIX (PSFH*PSFW*CC)       // 12675
#define KPIX (RPSF*RPSF*CC)       // 1587

typedef __attribute__((ext_vector_type(2))) float v2f;
typedef __attribute__((ext_vector_type(8))) float v8f;

#ifndef __has_builtin
#define __has_builtin(x) 0
#endif

__device__ __forceinline__ float field_r(int row, int col) {
    const float HD = 723.37027832f;   // hypot(1023,1023)/2
    float x = ((float)col - 511.5f) / HD;
    float y = ((float)row - 511.5f) / HD;
    return sqrtf(x * x + y * y);
}

__device__ __forceinline__ float lin_interp21(float soft, const float* v) {
    soft = fminf(fmaxf(soft, 0.f), 20.f);
    float fl = floorf(soft);
    int lo = (int)fl;
    int hi = (int)ceilf(soft);
    float f = soft - fl;
    return v[lo] * (1.f - f) + v[hi] * f;
}

__device__ __forceinline__ float cubw(float t) {  // Catmull-Rom a = -0.5
    float t2 = t * t, t3 = t2 * t;
    if (t <= 1.f) return 1.5f * t3 - 2.5f * t2 + 1.f;
    if (t < 2.f)  return -0.5f * t3 + 2.5f * t2 - 4.f * t + 2.f;
    return 0.f;
}

// ---------------- PSF_WEIGHTS: per-cell histogram of round(fieldmap*20) ----
__global__ __launch_bounds__(256) void k_weights(float* __restrict__ w_ws) {
    __shared__ int hist[NF];
    int g = blockIdx.x, t = threadIdx.x;
    if (t < NF) hist[t] = 0;
    __syncthreads();
    int gy = g >> 3, gx = g & 7;
    for (int i = t; i < 128 * 128; i += 256) {
        int row = gy * 128 + (i >> 7);
        int col = gx * 128 + (i & 127);
        float fm = fminf(field_r(row, col), 1.f);
        int d = __float2int_rn(fm * 20.f);  // round-nearest-even like jnp.round
        atomicAdd(&hist[d], 1);
    }
    __syncthreads();
    if (t < NF) w_ws[g * NF + t] = (float)hist[t] * (1.f / 16384.f);
}

// ---------------- interp = W(64x21) @ P(21x12675) via V_WMMA_F32_16X16X4 ---
__global__ __launch_bounds__(128) void k_interp(const float* __restrict__ W,
                                                const float* __restrict__ P,
                                                float* __restrict__ out) {
    int lane   = threadIdx.x & 31;
    int wave   = threadIdx.x >> 5;     // m-tile (16 cells each)
    int half   = lane >> 4;
    int lanelo = lane & 15;
    int n = blockIdx.x * 16 + lanelo;  // pixel-channel column
    v8f acc = {0.f, 0.f, 0.f, 0.f, 0.f, 0.f, 0.f, 0.f};
#if __has_builtin(__builtin_amdgcn_wmma_f32_16x16x4_f32)
    for (int ks = 0; ks < 6; ++ks) {   // K = 21 padded to 24
        int k0 = ks * 4;
        int ka = k0 + half * 2;        // lanes 0-15: K=k0,k0+1; lanes 16-31: K=k0+2,k0+3
        int m  = wave * 16 + lanelo;   // A rows: M = lane&15
        v2f a, b;
        a.x = (ka     < NF) ? W[m * NF + ka]     : 0.f;
        a.y = (ka + 1 < NF) ? W[m * NF + ka + 1] : 0.f;
        b.x = (ka     < NF && n < NPIX) ? P[ka * NPIX + n]       : 0.f;
        b.y = (ka + 1 < NF && n < NPIX) ? P[(ka + 1) * NPIX + n] : 0.f;
        acc = __builtin_amdgcn_wmma_f32_16x16x4_f32(false, a, false, b,
                                                    (short)0, acc, false, false);
    }
#else
    for (int k = 0; k < NF; ++k) {
        float pv = (n < NPIX) ? P[k * NPIX + n] : 0.f;
        #pragma unroll
        for (int i = 0; i < 8; ++i) {
            int m = wave * 16 + half * 8 + i;
            acc[i] += W[m * NF + k] * pv;
        }
    }
#endif
    if (n < NPIX) {
        #pragma unroll
        for (int i = 0; i < 8; ++i) {
            int m = wave * 16 + half * 8 + i;   // D rows: VGPR i -> M=i (+8 for hi half)
            out[(size_t)m * NPIX + n] = acc[i];
        }
    }
}

// ---------------- rotate (bilinear) + antialiased linear resize + normalize -
__global__ __launch_bounds__(256) void k_psf_finish(const float* __restrict__ interp,
                                                    float* __restrict__ psf_out) {
    __shared__ float sP[NPIX];    // 50.7 KB, fits in 320KB/WGP LDS
    __shared__ float sOut[KPIX];
    __shared__ float ssum[CC];
    int g = blockIdx.x, t = threadIdx.x;
    for (int i = t; i < NPIX; i += 256) sP[i] = interp[(size_t)g * NPIX + i];
    __syncthreads();
    int gy = g >> 3, gx = g & 7;
    float xc = ((float)gx + 0.5f) * (1023.f / 8.f) - 511.5f;
    float yc = ((float)gy + 0.5f) * (1023.f / 8.f) - 511.5f;
    float ang = -atan2f(xc, yc);                  // rotate by -ANGLES[g]
    float ca = cosf(ang), sa = sinf(ang);
    const float inv_scale = 65.f / 23.f;          // antialias kernel support
    const float scale = 23.f / 65.f;
    for (int p = t; p < RPSF * RPSF; p += 256) {
        int oy = p / RPSF, ox = p % RPSF;
        float sfy = ((float)oy + 0.5f) * inv_scale - 0.5f;
        float sfx = ((float)ox + 0.5f) * inv_scale - 0.5f;
        int y_lo = max(0, (int)ceilf(sfy - inv_scale));
        int y_hi = min(PSFH - 1, (int)floorf(sfy + inv_scale));
        int x_lo = max(0, (int)ceilf(sfx - inv_scale));
        int x_hi = min(PSFW - 1, (int)floorf(sfx + inv_scale));
        float wy[8], wx[8];
        float ysum = 0.f, xsum = 0.f;
        int ny = y_hi - y_lo + 1, nx = x_hi - x_lo + 1;
        for (int j = 0; j < ny; ++j) {
            float w = fmaxf(1.f - fabsf((float)(y_lo + j) - sfy) * scale, 0.f);
            wy[j] = w; ysum += w;
        }
        for (int j = 0; j < nx; ++j) {
            float w = fmaxf(1.f - fabsf((float)(x_lo + j) - sfx) * scale, 0.f);
            wx[j] = w; xsum += w;
        }
        float inv = 1.f / (ysum * xsum);
        float a0 = 0.f, a1 = 0.f, a2 = 0.f;
        for (int j = 0; j < ny; ++j) {
            int iy = y_lo + j;
            float dyy = (float)iy - 32.f;
            for (int i2 = 0; i2 < nx; ++i2) {
                float w = wy[j] * wx[i2];
                if (w == 0.f) continue;
                int ix = x_lo + i2;
                float dxx = (float)ix - 32.f;
                float sx = ca * dxx + sa * dyy + 32.f;
                float sy = -sa * dxx + ca * dyy + 32.f;
                float fxs = floorf(sx), fys = floorf(sy);
                int x0 = (int)fxs, y0 = (int)fys;
                float fx = sx - fxs, fy = sy - fys;
                float w00 = (1.f - fy) * (1.f - fx), w01 = (1.f - fy) * fx;
                float w10 = fy * (1.f - fx),         w11 = fy * fx;
                bool vy0 = (y0 >= 0) & (y0 < PSFH), vy1 = (y0 + 1 >= 0) & (y0 + 1 < PSFH);
                bool vx0 = (x0 >= 0) & (x0 < PSFW), vx1 = (x0 + 1 >= 0) & (x0 + 1 < PSFW);
                int b00 = (y0 * PSFW + x0) * CC;
                #pragma unroll
                for (int c = 0; c < CC; ++c) {
                    float v = 0.f;
                    if (vy0 & vx0) v += w00 * sP[b00 + c];
                    if (vy0 & vx1) v += w01 * sP[b00 + CC + c];
                    if (vy1 & vx0) v += w10 * sP[b00 + PSFW * CC + c];
                    if (vy1 & vx1) v += w11 * sP[b00 + (PSFW + 1) * CC + c];
                    float* acc = (c == 0) ? &a0 : (c == 1) ? &a1 : &a2;
                    *acc += w * v;
                }
            }
        }
        sOut[p * CC + 0] = a0 * inv;
        sOut[p * CC + 1] = a1 * inv;
        sOut[p * CC + 2] = a2 * inv;
    }
    __syncthreads();
    if (t < CC) {
        float s = 0.f;
        for (int p = 0; p < RPSF * RPSF; ++p) s += sOut[p * CC + t];
        ssum[t] = 1.f / s;
    }
    __syncthreads();
    for (int i = t; i < KPIX; i += 256)
        psf_out[(size_t)g * KPIX + i] = sOut[i] * ssum[i % CC];
}

// ---------------- spatially-varying depthwise 23x23 conv (RI fused) --------
// Thread -> 4 consecutive-x outputs; sliding 4-wide register window per channel
// gives 12 FMAs per 6 LDS loads per tap (VALU-bound with VOPD dual-issue).
__global__ __launch_bounds__(256) void k_conv(const float* __restrict__ img,
                                              const float* __restrict__ ri_vals,
                                              const float* __restrict__ psfs,
                                              float* __restrict__ blur) {
    __shared__ float sK[CC][RPSF * RPSF];
    __shared__ float sIn[CC][54 * 54];
    __shared__ float sRI[NF];
    int t = threadIdx.x;
    if (t < NF) sRI[t] = ri_vals[t];
    int bi   = blockIdx.x;
    int b    = bi & 3;
    int tile = (bi >> 2) & 15;
    int g    = bi >> 6;
    int gy = g >> 3, gx = g & 7;
    int row0c = gy * 128, col0c = gx * 128;
    int row0 = row0c + (tile >> 2) * 32;
    int col0 = col0c + (tile & 3) * 32;
    for (int i = t; i < KPIX; i += 256)
        sK[i % CC][i / CC] = psfs[(size_t)g * KPIX + i];
    __syncthreads();
    for (int i = t; i < 54 * 54; i += 256) {
        int iy = i / 54, ix = i % 54;
        int row = row0 - 11 + iy;
        int col = col0 - 11 + ix;
        float v0 = 0.f, v1 = 0.f, v2 = 0.f;
        // SAME conv is per-patch: zero outside this cell's 128x128 patch
        if (row >= row0c && row < row0c + 128 && col >= col0c && col < col0c + 128) {
            float fm = fminf(field_r(row, col), 1.f);
            float ri = lin_interp21(fm * 20.f, sRI);
            const float* p = img + (((size_t)b * HIMG + row) * WIMG + col) * CC;
            v0 = p[0] * ri; v1 = p[1] * ri; v2 = p[2] * ri;
        }
        sIn[0][i] = v0; sIn[1][i] = v1; sIn[2][i] = v2;
    }
    __syncthreads();
    int oy  = t >> 3;          // 0..31
    int ox0 = (t & 7) * 4;     // 0,4,...,28
    float acc[4][CC] = {};     // [x-offset j][channel]
    for (int dy = 0; dy < RPSF; ++dy) {
        int irow = (oy + dy) * 54 + ox0;
        int krow = dy * RPSF;
        float r0[CC], r1[CC], r2[CC], r3[CC];
        #pragma unroll
        for (int c = 0; c < CC; ++c) {
            r0[c] = sIn[c][irow + 0];
            r1[c] = sIn[c][irow + 1];
            r2[c] = sIn[c][irow + 2];
        }
        #pragma unroll
        for (int dx = 0; dx < RPSF; ++dx) {
            #pragma unroll
            for (int c = 0; c < CC; ++c) {
                r3[c] = sIn[c][irow + dx + 3];
                float k = sK[c][krow + dx];
                acc[0][c] = fmaf(r0[c], k, acc[0][c]);
                acc[1][c] = fmaf(r1[c], k, acc[1][c]);
                acc[2][c] = fmaf(r2[c], k, acc[2][c]);
                acc[3][c] = fmaf(r3[c], k, acc[3][c]);
                r0[c] = r1[c]; r1[c] = r2[c]; r2[c] = r3[c];
            }
        }
    }
    int row = row0 + oy;
    float* o = blur + (((size_t)b * HIMG + row) * WIMG + (col0 + ox0)) * CC;
    #pragma unroll
    for (int j = 0; j < 4; ++j) {
        o[j * CC + 0] = acc[j][0];
        o[j * CC + 1] = acc[j][1];
        o[j * CC + 2] = acc[j][2];
    }
}

// ---------------- bicubic distortion warp ----------------------------------
__global__ __launch_bounds__(256) void k_warp(const float* __restrict__ blur,
                                              const float* __restrict__ shifts,
                                              float* __restrict__ out) {
    int pid = blockIdx.x * 256 + threadIdx.x;
    int y = pid >> 10, x = pid & 1023;
    const float HD = 723.37027832f;
    float xm = ((float)x - 511.5f) / HD;
    float ym = ((float)y - 511.5f) / HD;
    float r = sqrtf(xm * xm + ym * ym);
    float angp = atan2f(ym, xm);
    float soft = fminf(fmaxf(r * 20.f, 0.f), 20.f);
    float fl = floorf(soft);
    int lo = (int)fl, hi = (int)ceilf(soft);
    float f = soft - fl;
    float shift = shifts[lo] * (1.f - f) + shifts[hi] * f;
    float qx = (float)x + shift * cosf(angp) * HD;
    float qy = (float)y + shift * sinf(angp) * HD;
    float fqx = floorf(qx), fqy = floorf(qy);
    int x0 = (int)fqx, y0 = (int)fqy;
    float fx = qx - fqx, fy = qy - fqy;
    float wx[4], wy[4];
    #pragma unroll
    for (int i = 0; i < 4; ++i) {
        wx[i] = cubw(fabsf(fx - (float)(i - 1)));
        wy[i] = cubw(fabsf(fy - (float)(i - 1)));
    }
    float acc[BB][CC] = {};
    #pragma unroll
    for (int j = 0; j < 4; ++j) {
        int yj = min(max(y0 + j - 1, 0), HIMG - 1);
        #pragma unroll
        for (int i = 0; i < 4; ++i) {
            int xi = min(max(x0 + i - 1, 0), WIMG - 1);
            float w = wy[j] * wx[i];
            #pragma unroll
            for (int b = 0; b < BB; ++b) {
                const float* p = blur + (((size_t)b * HIMG + yj) * WIMG + xi) * CC;
                acc[b][0] += w * p[0];
                acc[b][1] += w * p[1];
                acc[b][2] += w * p[2];
            }
        }
    }
    #pragma unroll
    for (int b = 0; b < BB; ++b) {
        float* o = out + (((size_t)b * HIMG + y) * WIMG + x) * CC;
        o[0] = acc[b][0]; o[1] = acc[b][1]; o[2] = acc[b][2];
    }
}

extern "C" void kernel_launch(void* const* d_in, const int* in_sizes, int n_in,
                              void* d_out, int out_size, void* d_ws, size_t ws_size,
                              hipStream_t stream) {
    const float* img      = (const float*)d_in[0];  // (4,1024,1024,3)
    const float* psfs_in  = (const float*)d_in[1];  // (21,65,65,3)
    const float* shifts   = (const float*)d_in[2];  // (21,)
    const float* ri       = (const float*)d_in[3];  // (21,)
    float* out = (float*)d_out;

    float* w_ws      = (float*)d_ws;                 // 64*21
    float* interp_ws = w_ws + 2048;                  // 64*12675
    float* psf_ws    = interp_ws + 811520;           // 64*1587
    float* blur_ws   = psf_ws + 102400;              // 4*1024*1024*3

    hipLaunchKernelGGL(k_weights,    dim3(64),   dim3(256), 0, stream, w_ws);
    hipLaunchKernelGGL(k_interp,     dim3(793),  dim3(128), 0, stream, w_ws, psfs_in, interp_ws);
    hipLaunchKernelGGL(k_psf_finish, dim3(64),   dim3(256), 0, stream, interp_ws, psf_ws);
    hipLaunchKernelGGL(k_conv,       dim3(4096), dim3(256), 0, stream, img, ri, psf_ws, blur_ws);
    hipLaunchKernelGGL(k_warp,       dim3(4096), dim3(256), 0, stream, blur_ws, shifts, out);
}